// HeteroSAGE_24773371363384
// MI455X (gfx1250) — compile-verified
//
#include <hip/hip_runtime.h>
#include <math.h>

// ---------------- problem constants (from reference) ----------------
#define NNODES 50000
#define NEDGES 600000
#define DIM    128      // D == H == 128
#define NCLS   40
#define NCLS_PAD 48     // decoder weight columns padded to multiple of 16
#define NEG_SLOPE 0.1f

typedef __attribute__((ext_vector_type(2))) float v2f;
typedef __attribute__((ext_vector_type(4))) float v4f;
typedef __attribute__((ext_vector_type(8))) float v8f;

// ---------------- helpers ----------------
__device__ __forceinline__ void atomAddF(float* p, float v) {
    // agent-scope relaxed fp32 add -> global_atomic_add_f32 on gfx1250
    __hip_atomic_fetch_add(p, v, __ATOMIC_RELAXED, __HIP_MEMORY_SCOPE_AGENT);
}

// ---------------- zero fill (grid-stride) ----------------
__global__ void zero_kernel(float* __restrict__ p, int n) {
    for (int i = blockIdx.x * blockDim.x + threadIdx.x; i < n;
         i += gridDim.x * blockDim.x)
        p[i] = 0.0f;
}

// ---------------- degree count + invert ----------------
__global__ void degree_kernel(const int* __restrict__ dst, float* __restrict__ deg) {
    int e = blockIdx.x * blockDim.x + threadIdx.x;
    if (e < NEDGES) atomAddF(&deg[dst[e]], 1.0f);
}

__global__ void invert_kernel(float* __restrict__ deg) {
    int i = blockIdx.x * blockDim.x + threadIdx.x;
    if (i < NNODES) deg[i] = 1.0f / fmaxf(deg[i], 1.0f);
}

// ---------------- weight transpose (+ zero pad) ----------------
// W is [DIM x ncols] row-major; WT is [npad x DIM] row-major with WT[n][k] =
// (n < ncols) ? W[k][n] : 0.  Makes B fragments contiguous b64 loads.
__global__ void transpose_pad_kernel(const float* __restrict__ W,
                                     float* __restrict__ WT,
                                     int ncols, int npad) {
    int t = blockIdx.x * blockDim.x + threadIdx.x;   // t = n*DIM + k
    if (t >= npad * DIM) return;
    int n = t >> 7;          // / DIM
    int k = t & (DIM - 1);   // % DIM
    WT[t] = (n < ncols) ? W[k * ncols + n] : 0.0f;
}

// ---------------- edge scatter-add: agg[dst] += x[src] ----------------
// one thread per (edge, 4-float chunk): E * 32 chunks of float4
__global__ void scatter_kernel(const float* __restrict__ xin,
                               const int* __restrict__ src,
                               const int* __restrict__ dst,
                               float* __restrict__ agg) {
    int t = blockIdx.x * blockDim.x + threadIdx.x;
    if (t >= NEDGES * 32) return;
    int e = t >> 5;
    int c = (t & 31) << 2;
    int s = src[e];
    int d = dst[e];
    v4f v = *(const v4f*)(xin + (size_t)s * DIM + c);
    float* p = agg + (size_t)d * DIM + c;
    atomAddF(p + 0, v.x);
    atomAddF(p + 1, v.y);
    atomAddF(p + 2, v.z);
    atomAddF(p + 3, v.w);
}

// ---------------- fused SAGE GEMM with fp32 WMMA ----------------
// out[M,DOUT] = (agg*inv_cnt) @ Wl + xin @ Wr + bias, optional leaky-relu.
// One wave32 computes one 16x16 output tile with V_WMMA_F32_16X16X4_F32.
// A frag (16x4 f32): lane holds row m = lane&15; VGPR0/1 = K = kb, kb+1,
//   kb = 2*(lane>>4).  B frag (4x16): lane holds col n = lane&15, same K pair;
//   loaded from transposed weights WT[n][k] as one contiguous b64.
// C/D (16x16 f32, 8 VGPRs): row = v + 8*(lane>>4), col = lane&15.
template <int NPAD, int DOUT, bool RELU>
__global__ __launch_bounds__(32)
void sage_gemm_kernel(const float* __restrict__ agg,
                      const float* __restrict__ xin,
                      const float* __restrict__ invc,
                      const float* __restrict__ WlT,   // [NPAD][DIM]
                      const float* __restrict__ WrT,   // [NPAD][DIM]
                      const float* __restrict__ bias,  // [DOUT]
                      float* __restrict__ out) {
    const int lane  = threadIdx.x & 31;
    const int half  = lane >> 4;      // selects K pair (0 -> K0,K1 ; 1 -> K2,K3)
    const int idx16 = lane & 15;      // A row within tile / B,C column within tile
    const int mbase = blockIdx.x * 16;
    const int nbase = blockIdx.y * 16;

    const int arow = mbase + idx16;   // NNODES is a multiple of 16
    const int bcol = nbase + idx16;   // < NPAD by construction
    const int kb   = half * 2;

    const float  scale = invc[arow];
    const float* Ar = agg + arow * DIM + kb;
    const float* Xr = xin + arow * DIM + kb;
    const float* Bl = WlT + bcol * DIM + kb;
    const float* Br = WrT + bcol * DIM + kb;

    v8f acc = {};

    // agg * inv_cnt  @  Wl
#pragma unroll
    for (int k = 0; k < DIM; k += 4) {
        v2f a = *(const v2f*)(Ar + k);
        a.x *= scale;
        a.y *= scale;
        v2f b = *(const v2f*)(Bl + k);
        acc = __builtin_amdgcn_wmma_f32_16x16x4_f32(false, a, false, b,
                                                    (short)0, acc, false, false);
    }
    // xin @ Wr (same accumulator)
#pragma unroll
    for (int k = 0; k < DIM; k += 4) {
        v2f a = *(const v2f*)(Xr + k);
        v2f b = *(const v2f*)(Br + k);
        acc = __builtin_amdgcn_wmma_f32_16x16x4_f32(false, a, false, b,
                                                    (short)0, acc, false, false);
    }

    const bool colok = (NPAD == DOUT) || (bcol < DOUT);   // constant-true for 128
    if (colok) {
        const float bv = bias[bcol];
#pragma unroll
        for (int v = 0; v < 8; ++v) {
            int   r   = mbase + v + 8 * half;
            float val = acc[v] + bv;
            if (RELU) val = (val > 0.0f) ? val : NEG_SLOPE * val;
            out[r * DOUT + bcol] = val;
        }
    }
}

// ---------------- log-softmax over rows of 40, one wave32 per row ----------
__global__ __launch_bounds__(256)
void logsoftmax_kernel(float* __restrict__ out) {
    const int wave = threadIdx.x >> 5;
    const int lane = threadIdx.x & 31;
    const int row  = blockIdx.x * 8 + wave;
    if (row >= NNODES) return;   // uniform per wave

    float* p = out + (size_t)row * NCLS;
    float v0 = (lane < NCLS)      ? p[lane]      : -__builtin_inff();
    float v1 = (32 + lane < NCLS) ? p[32 + lane] : -__builtin_inff();

    float m = fmaxf(v0, v1);
#pragma unroll
    for (int off = 16; off > 0; off >>= 1)
        m = fmaxf(m, __shfl_xor(m, off, 32));

    float s = ((lane < NCLS) ? expf(v0 - m) : 0.0f) +
              ((32 + lane < NCLS) ? expf(v1 - m) : 0.0f);
#pragma unroll
    for (int off = 16; off > 0; off >>= 1)
        s += __shfl_xor(s, off, 32);

    const float lse = m + logf(s);
    if (lane < NCLS)      p[lane]      = v0 - lse;
    if (32 + lane < NCLS) p[32 + lane] = v1 - lse;
}

// ---------------- launcher ----------------
extern "C" void kernel_launch(void* const* d_in, const int* in_sizes, int n_in,
                              void* d_out, int out_size, void* d_ws, size_t ws_size,
                              hipStream_t stream) {
    (void)in_sizes; (void)n_in; (void)out_size; (void)ws_size;

    const float* x      = (const float*)d_in[0];
    const int*   ei     = (const int*)  d_in[1];
    const int*   src    = ei;
    const int*   dst    = ei + NEDGES;
    const float* enc_Wl = (const float*)d_in[2];
    const float* enc_Wr = (const float*)d_in[3];
    const float* enc_b  = (const float*)d_in[4];
    const float* lay_Wl = (const float*)d_in[5];   // [L, H, H]
    const float* lay_Wr = (const float*)d_in[6];
    const float* lay_b  = (const float*)d_in[7];   // [L, H]
    const float* dec_Wl = (const float*)d_in[8];
    const float* dec_Wr = (const float*)d_in[9];
    const float* dec_b  = (const float*)d_in[10];
    float* outp = (float*)d_out;

    float* ws  = (float*)d_ws;
    float* inv = ws;                                  // N
    float* agg = inv + NNODES;                        // N*DIM
    float* hA  = agg + (size_t)NNODES * DIM;          // N*DIM
    float* hB  = hA  + (size_t)NNODES * DIM;          // N*DIM
    float* wt  = hB  + (size_t)NNODES * DIM;          // transposed weights
    float* encT_Wl  = wt;                 // 128*128
    float* encT_Wr  = encT_Wl + DIM*DIM;
    float* lay0T_Wl = encT_Wr + DIM*DIM;
    float* lay0T_Wr = lay0T_Wl + DIM*DIM;
    float* lay1T_Wl = lay0T_Wr + DIM*DIM;
    float* lay1T_Wr = lay1T_Wl + DIM*DIM;
    float* decT_Wl  = lay1T_Wr + DIM*DIM; // 48*128, zero padded
    float* decT_Wr  = decT_Wl + NCLS_PAD*DIM;

    // one-time weight transposes (tiny)
    const int tpH = (DIM * DIM + 255) / 256;        // 128x128
    const int tpC = (NCLS_PAD * DIM + 255) / 256;   // 48x128
    transpose_pad_kernel<<<tpH, 256, 0, stream>>>(enc_Wl, encT_Wl, DIM, DIM);
    transpose_pad_kernel<<<tpH, 256, 0, stream>>>(enc_Wr, encT_Wr, DIM, DIM);
    transpose_pad_kernel<<<tpH, 256, 0, stream>>>(lay_Wl,             lay0T_Wl, DIM, DIM);
    transpose_pad_kernel<<<tpH, 256, 0, stream>>>(lay_Wr,             lay0T_Wr, DIM, DIM);
    transpose_pad_kernel<<<tpH, 256, 0, stream>>>(lay_Wl + DIM * DIM, lay1T_Wl, DIM, DIM);
    transpose_pad_kernel<<<tpH, 256, 0, stream>>>(lay_Wr + DIM * DIM, lay1T_Wr, DIM, DIM);
    transpose_pad_kernel<<<tpC, 256, 0, stream>>>(dec_Wl, decT_Wl, NCLS, NCLS_PAD);
    transpose_pad_kernel<<<tpC, 256, 0, stream>>>(dec_Wr, decT_Wr, NCLS, NCLS_PAD);

    // inverse in-degree (clamped to 1)
    zero_kernel<<<256, 256, 0, stream>>>(inv, NNODES);
    degree_kernel<<<(NEDGES + 255) / 256, 256, 0, stream>>>(dst, inv);
    invert_kernel<<<(NNODES + 255) / 256, 256, 0, stream>>>(inv);

    const int scat_blocks = (NEDGES * 32 + 255) / 256;
    auto aggregate = [&](const float* in) {
        zero_kernel<<<2048, 256, 0, stream>>>(agg, NNODES * DIM);
        scatter_kernel<<<scat_blocks, 256, 0, stream>>>(in, src, dst, agg);
    };

    const dim3 gridH(NNODES / 16, DIM / 16);
    const dim3 gridC(NNODES / 16, NCLS_PAD / 16);

    // encoder D->H
    aggregate(x);
    sage_gemm_kernel<DIM, DIM, true><<<gridH, 32, 0, stream>>>(
        agg, x, inv, encT_Wl, encT_Wr, enc_b, hA);
    // layer 0
    aggregate(hA);
    sage_gemm_kernel<DIM, DIM, true><<<gridH, 32, 0, stream>>>(
        agg, hA, inv, lay0T_Wl, lay0T_Wr, lay_b, hB);
    // layer 1
    aggregate(hB);
    sage_gemm_kernel<DIM, DIM, true><<<gridH, 32, 0, stream>>>(
        agg, hB, inv, lay1T_Wl, lay1T_Wr, lay_b + DIM, hA);
    // decoder H->C (no activation)
    aggregate(hA);
    sage_gemm_kernel<NCLS_PAD, NCLS, false><<<gridC, 32, 0, stream>>>(
        agg, hA, inv, decT_Wl, decT_Wr, dec_b, outp);

    // row-wise log-softmax on d_out
    logsoftmax_kernel<<<NNODES / 8, 256, 0, stream>>>(outp);
}